// CategoricalEntropyRegLoss_27032524161204
// MI455X (gfx1250) — compile-verified
//
#include <hip/hip_runtime.h>
#include <hip/hip_bf16.h>
#include <math.h>

// ---------------------------------------------------------------------------
// CategoricalEntropyRegLoss on MI455X (gfx1250, wave32, WMMA fp32 16x16x4)
// B=2048, F=1024, D=8, C=32, DC=256.
//  - All GEMMs via V_WMMA_F32_16X16X4_F32 (fp32 kept: loss built from logs/KL)
//  - Fused diversity kernel: Gram(fn) + P@L^T + L@P^T -> scalar, no BxB mats
//  - Double-buffered LDS staging via GLOBAL_LOAD_ASYNC_TO_LDS_B128 (ASYNCcnt)
// ---------------------------------------------------------------------------

typedef __attribute__((ext_vector_type(2))) float v2f;
typedef __attribute__((ext_vector_type(8))) float v8f;
typedef int v4i __attribute__((vector_size(16)));   // matches builtin's V4i
typedef __attribute__((address_space(1))) v4i* gv4i_p;
typedef __attribute__((address_space(3))) v4i* lv4i_p;

#define B_SZ   2048
#define F_SZ   1024
#define D_SZ   8
#define DC_SZ  256
#define KTILE  64
#define LSTR   68   // LDS row stride (floats) for KTILE=64 tiles
#define KTILE2 32
#define LSTR2  36   // LDS row stride (floats) for KTILE=32 tiles
#define MSTR   36   // LDS stride for centers A-tile (32 cols + pad)

// Workspace layout (float offsets)
#define OFF_FN       ((size_t)0)                          // 2048*1024
#define OFF_P        (OFF_FN + (size_t)B_SZ * F_SZ)       // 2048*256
#define OFF_LP       (OFF_P  + (size_t)B_SZ * DC_SZ)      // 2048*256
#define OFF_TM       (OFF_LP + (size_t)B_SZ * DC_SZ)      // 2048*256 (targets*mask)
#define OFF_CENTERS  (OFF_TM + (size_t)B_SZ * DC_SZ)      // 256*1024
#define OFF_SQ       (OFF_CENTERS + (size_t)DC_SZ * F_SZ) // 2048
#define OFF_E        (OFF_SQ + B_SZ)                      // 2048
#define OFF_WSUM     (OFF_E + B_SZ)                       // 256
#define OFF_CSQ      (OFF_WSUM + DC_SZ)                   // 256
#define OFF_VALID    (OFF_CSQ + DC_SZ)                    // 256
#define OFF_SCAL     (OFF_VALID + DC_SZ)                  // 4: [S_div, S_tight, M, pad]

// ---- async global->LDS path (guarded: falls back to sync copy) ------------
#if defined(__has_builtin)
#if __has_builtin(__builtin_amdgcn_global_load_async_to_lds_b128) && \
    __has_builtin(__builtin_amdgcn_s_wait_asynccnt)
#define HAVE_ASYNC_LDS 1
#endif
#endif
#ifndef HAVE_ASYNC_LDS
#define HAVE_ASYNC_LDS 0
#endif

__device__ __forceinline__ void async_copy16(const float* g, float* l) {
#if HAVE_ASYNC_LDS
  // per-lane 16B DMA into LDS, tracked by ASYNCcnt (no VGPR round-trip).
  // Generic->AS casts via integer round-trip; LDS offset = low 32 addr bits.
  __builtin_amdgcn_global_load_async_to_lds_b128(
      (gv4i_p)(unsigned long long)g,
      (lv4i_p)(unsigned int)(unsigned long long)l,
      0, 0);
#else
  *(float4*)l = *(const float4*)g;
#endif
}

__device__ __forceinline__ void wait_async() {
#if HAVE_ASYNC_LDS
  __builtin_amdgcn_s_wait_asynccnt(0);
#endif
}

// copy rows x (cols4*4) floats, 16B granules; cols4 is a compile-time const
__device__ __forceinline__ void stage_block(const float* __restrict__ g,
                                            size_t gstride, float* lds, int lstr,
                                            int rows, int cols4) {
  const int tid = threadIdx.x;
  const int total = rows * cols4;
  for (int idx = tid; idx < total; idx += 256) {
    const int r = idx / cols4;
    const int c4 = (idx - r * cols4) << 2;
    async_copy16(g + (size_t)r * gstride + c4, lds + r * lstr + c4);
  }
}

__device__ __forceinline__ float waveReduceSum(float v) {
  #pragma unroll
  for (int off = 16; off > 0; off >>= 1) v += __shfl_xor(v, off, 32);
  return v;
}

__device__ __forceinline__ v8f wmma_f32(v2f a, v2f b, v8f c) {
  return __builtin_amdgcn_wmma_f32_16x16x4_f32(
      false, a, false, b, (short)0, c, false, false);
}

// ---------------------------------------------------------------------------
__global__ void zero_kernel(float* scal) {
  scal[0] = 0.f; scal[1] = 0.f; scal[2] = 0.f; scal[3] = 0.f;
}

// fn = features / max(||features||,1e-12); sq[b] = sum(fn^2)
__global__ __launch_bounds__(256)
void normalize_kernel(const float* __restrict__ features,
                      float* __restrict__ fn, float* __restrict__ sq) {
  __shared__ float red[8];
  const int b = blockIdx.x, tid = threadIdx.x;
  const float4 v = *(const float4*)&features[(size_t)b * F_SZ + tid * 4];
  float s = v.x*v.x + v.y*v.y + v.z*v.z + v.w*v.w;
  s = waveReduceSum(s);
  if ((tid & 31) == 0) red[tid >> 5] = s;
  __syncthreads();
  float tot = 0.f;
  #pragma unroll
  for (int i = 0; i < 8; ++i) tot += red[i];
  const float inv = 1.f / fmaxf(sqrtf(tot), 1e-12f);
  float4 o; o.x = v.x*inv; o.y = v.y*inv; o.z = v.z*inv; o.w = v.w*inv;
  *(float4*)&fn[(size_t)b * F_SZ + tid * 4] = o;
  if (tid == 0) sq[b] = tot * inv * inv;
}

// p = (t+eps)/rowsum, logp, E[b] = sum_d sum_c p*logp, tm = targets*mask
// One wave per (b,d): 32 lanes == 32 classes.
__global__ __launch_bounds__(256)
void prep_kl_kernel(const float* __restrict__ targets,
                    const unsigned char* __restrict__ mask,
                    float* __restrict__ p, float* __restrict__ lp,
                    float* __restrict__ tm, float* __restrict__ E) {
  __shared__ float entd[8];
  const int b = blockIdx.x;
  const int d = threadIdx.x >> 5, c = threadIdx.x & 31;
  const size_t off = (size_t)b * DC_SZ + d * 32 + c;
  const float traw = targets[off];
  tm[off] = traw * (mask[b] ? 1.f : 0.f);
  const float t = traw + 1e-10f;
  const float s = waveReduceSum(t);
  const float pv = t / s;
  const float lpv = logf(pv);
  p[off] = pv;
  lp[off] = lpv;
  const float e = waveReduceSum(pv * lpv);
  if (c == 0) entd[d] = e;
  __syncthreads();
  if (threadIdx.x == 0) {
    float es = 0.f;
    #pragma unroll
    for (int dd = 0; dd < 8; ++dd) es += entd[dd];
    E[b] = es;
  }
}

// wsum[dc] = sum_b tm[b,dc];  M = sum_b m[b]
__global__ __launch_bounds__(256)
void wsum_kernel(const float* __restrict__ tm,
                 const unsigned char* __restrict__ mask,
                 float* __restrict__ wsum, float* __restrict__ Mout) {
  const int t = threadIdx.x;
  float s = 0.f;
  for (int b = 0; b < B_SZ; ++b) s += tm[(size_t)b * DC_SZ + t];
  wsum[t] = s;
  float mloc = 0.f;
  for (int b = t; b < B_SZ; b += 256) mloc += mask[b] ? 1.f : 0.f;
  __shared__ float red[8];
  mloc = waveReduceSum(mloc);
  if ((t & 31) == 0) red[t >> 5] = mloc;
  __syncthreads();
  if (t == 0) {
    float M = 0.f;
    #pragma unroll
    for (int i = 0; i < 8; ++i) M += red[i];
    *Mout = M;
  }
}

// ---------------------------------------------------------------------------
// Fused diversity kernel (32x64 tile / 8 waves of 16x16 WMMA):
//   G = fn_i.fn_j (K=1024); cIJ = p_i.logp_j, cJI = logp_i.p_j (K=256)
//   S_div += (sq_i+sq_j-2G) * (0.5/D)(E_i+E_j-cIJ-cJI) * m_i m_j [i!=j]
// ---------------------------------------------------------------------------
#define H1 (96 * LSTR)    // phase-1 buffer half (32+64 rows)
#define H2 (192 * LSTR2)  // phase-2 buffer half (4 * (32|64) rows)

__global__ __launch_bounds__(256)
void diversity_kernel(const float* __restrict__ fn, const float* __restrict__ p,
                      const float* __restrict__ lp, const float* __restrict__ sq,
                      const float* __restrict__ E,
                      const unsigned char* __restrict__ mask,
                      float* __restrict__ S_div) {
  __shared__ __align__(16) float smem[2 * H2];  // >= 2*H1 as well
  const int tid  = threadIdx.x;
  const int wave = tid >> 5, lane = tid & 31;
  const int hf   = lane >> 4, l16 = lane & 15;
  const int wr   = wave >> 2, wc  = wave & 3;
  const int i0   = blockIdx.y * 32, j0 = blockIdx.x * 64;

  v8f accG = {}, accIJ = {}, accJI = {};

  // ---- Phase 1: Gram over fn, K=1024, double-buffered ----
  {
    int buf = 0;
    stage_block(fn + (size_t)i0 * F_SZ, F_SZ, smem, LSTR, 32, 16);
    stage_block(fn + (size_t)j0 * F_SZ, F_SZ, smem + 32 * LSTR, LSTR, 64, 16);
    wait_async();
    __syncthreads();
    for (int k0 = 0; k0 < F_SZ; k0 += KTILE) {
      float* cur = smem + buf * H1;
      if (k0 + KTILE < F_SZ) {
        float* nxt = smem + (buf ^ 1) * H1;
        stage_block(fn + (size_t)i0 * F_SZ + k0 + KTILE, F_SZ, nxt, LSTR, 32, 16);
        stage_block(fn + (size_t)j0 * F_SZ + k0 + KTILE, F_SZ, nxt + 32 * LSTR, LSTR, 64, 16);
      }
      const float* Ai = cur;
      const float* Bj = cur + 32 * LSTR;
      #pragma unroll
      for (int kk = 0; kk < KTILE; kk += 4) {
        const v2f a = *(const v2f*)&Ai[(16 * wr + l16) * LSTR + kk + 2 * hf];
        const v2f b = *(const v2f*)&Bj[(16 * wc + l16) * LSTR + kk + 2 * hf];
        accG = wmma_f32(a, b, accG);
      }
      wait_async();
      __syncthreads();
      buf ^= 1;
    }
  }

  // ---- Phase 2: KL cross terms, K=256 (flattened D*C), double-buffered ----
  {
    int buf = 0;
    stage_block(p  + (size_t)i0 * DC_SZ, DC_SZ, smem,               LSTR2, 32, 8);
    stage_block(lp + (size_t)i0 * DC_SZ, DC_SZ, smem + 32 * LSTR2,  LSTR2, 32, 8);
    stage_block(p  + (size_t)j0 * DC_SZ, DC_SZ, smem + 64 * LSTR2,  LSTR2, 64, 8);
    stage_block(lp + (size_t)j0 * DC_SZ, DC_SZ, smem + 128 * LSTR2, LSTR2, 64, 8);
    wait_async();
    __syncthreads();
    for (int k0 = 0; k0 < DC_SZ; k0 += KTILE2) {
      float* cur = smem + buf * H2;
      if (k0 + KTILE2 < DC_SZ) {
        float* nxt = smem + (buf ^ 1) * H2;
        const int kn = k0 + KTILE2;
        stage_block(p  + (size_t)i0 * DC_SZ + kn, DC_SZ, nxt,               LSTR2, 32, 8);
        stage_block(lp + (size_t)i0 * DC_SZ + kn, DC_SZ, nxt + 32 * LSTR2,  LSTR2, 32, 8);
        stage_block(p  + (size_t)j0 * DC_SZ + kn, DC_SZ, nxt + 64 * LSTR2,  LSTR2, 64, 8);
        stage_block(lp + (size_t)j0 * DC_SZ + kn, DC_SZ, nxt + 128 * LSTR2, LSTR2, 64, 8);
      }
      const float* Pi = cur;
      const float* Li = cur + 32 * LSTR2;
      const float* Pj = cur + 64 * LSTR2;
      const float* Lj = cur + 128 * LSTR2;
      #pragma unroll
      for (int kk = 0; kk < KTILE2; kk += 4) {
        const int ao = (16 * wr + l16) * LSTR2 + kk + 2 * hf;
        const int bo = (16 * wc + l16) * LSTR2 + kk + 2 * hf;
        const v2f pa = *(const v2f*)&Pi[ao];
        const v2f la = *(const v2f*)&Li[ao];
        const v2f pb = *(const v2f*)&Pj[bo];
        const v2f lb = *(const v2f*)&Lj[bo];
        accIJ = wmma_f32(pa, lb, accIJ);
        accJI = wmma_f32(la, pb, accJI);
      }
      wait_async();
      __syncthreads();
      buf ^= 1;
    }
  }

  // ---- Epilogue: fold into scalar ----
  float local = 0.f;
  #pragma unroll
  for (int v = 0; v < 8; ++v) {
    const int i = i0 + 16 * wr + v + 8 * hf;   // C/D layout: M = v + 8*(lane>=16)
    const int j = j0 + 16 * wc + l16;          //             N = lane&15
    const float fd = sq[i] + sq[j] - 2.f * accG[v];
    const float td = 0.0625f * (E[i] + E[j] - accIJ[v] - accJI[v]); // 0.5/D
    const float w  = (mask[i] && mask[j] && (i != j)) ? 1.f : 0.f;
    local += fd * td * w;
  }
  local = waveReduceSum(local);
  if (lane == 0) atomicAdd(S_div, local);
}

// ---------------------------------------------------------------------------
// centers_raw[m,f] = sum_k tm[k,m] * fn[k,f]   (K = B = 2048)
// A-tile staged K-major ([k][m], stride MSTR) so it is a pure 16B copy of tm.
// ---------------------------------------------------------------------------
#define HC (64 * MSTR + 64 * LSTR)  // buffer half: At(64x36) + Bt(64x68)

__global__ __launch_bounds__(256)
void centers_kernel(const float* __restrict__ tm, const float* __restrict__ fn,
                    float* __restrict__ centers) {
  __shared__ __align__(16) float smem[2 * HC];
  const int tid  = threadIdx.x;
  const int wave = tid >> 5, lane = tid & 31;
  const int hf   = lane >> 4, l16 = lane & 15;
  const int wr   = wave >> 2, wc  = wave & 3;
  const int m0   = blockIdx.y * 32, f0 = blockIdx.x * 64;

  v8f acc = {};
  int buf = 0;
  stage_block(tm + m0, DC_SZ, smem, MSTR, 64, 8);
  stage_block(fn + f0, F_SZ, smem + 64 * MSTR, LSTR, 64, 16);
  wait_async();
  __syncthreads();
  for (int k0 = 0; k0 < B_SZ; k0 += KTILE) {
    float* cur = smem + buf * HC;
    if (k0 + KTILE < B_SZ) {
      float* nxt = smem + (buf ^ 1) * HC;
      const int kn = k0 + KTILE;
      stage_block(tm + (size_t)kn * DC_SZ + m0, DC_SZ, nxt, MSTR, 64, 8);
      stage_block(fn + (size_t)kn * F_SZ + f0, F_SZ, nxt + 64 * MSTR, LSTR, 64, 16);
    }
    const float* At = cur;              // [k][m], stride MSTR
    const float* Bt = cur + 64 * MSTR;  // [k][n], stride LSTR
    #pragma unroll
    for (int kk = 0; kk < KTILE; kk += 4) {
      v2f a, b;
      a.x = At[(kk + 2 * hf)     * MSTR + 16 * wr + l16];
      a.y = At[(kk + 2 * hf + 1) * MSTR + 16 * wr + l16];
      b.x = Bt[(kk + 2 * hf)     * LSTR + 16 * wc + l16];
      b.y = Bt[(kk + 2 * hf + 1) * LSTR + 16 * wc + l16];
      acc = wmma_f32(a, b, acc);
    }
    wait_async();
    __syncthreads();
    buf ^= 1;
  }
  #pragma unroll
  for (int v = 0; v < 8; ++v) {
    const int m = m0 + 16 * wr + v + 8 * hf;
    const int f = f0 + 16 * wc + l16;
    centers[(size_t)m * F_SZ + f] = acc[v];
  }
}

// centers /= max(wsum,1e-30);  csq = ||center||^2;  valid = wsum>0
__global__ __launch_bounds__(256)
void centers_norm_kernel(float* __restrict__ centers,
                         const float* __restrict__ wsum,
                         float* __restrict__ csq, float* __restrict__ validf) {
  __shared__ float red[8];
  const int dc = blockIdx.x, t = threadIdx.x;
  const float w = wsum[dc];
  const float inv = 1.f / fmaxf(w, 1e-30f);
  float4 v = *(float4*)&centers[(size_t)dc * F_SZ + t * 4];
  v.x *= inv; v.y *= inv; v.z *= inv; v.w *= inv;
  *(float4*)&centers[(size_t)dc * F_SZ + t * 4] = v;
  float s = v.x*v.x + v.y*v.y + v.z*v.z + v.w*v.w;
  s = waveReduceSum(s);
  if ((t & 31) == 0) red[t >> 5] = s;
  __syncthreads();
  if (t == 0) {
    float tot = 0.f;
    #pragma unroll
    for (int i = 0; i < 8; ++i) tot += red[i];
    csq[dc] = tot;
    validf[dc] = (w > 0.f) ? 1.f : 0.f;
  }
}

// ---------------------------------------------------------------------------
// tightness: dot[b,dc] = fn_b . center_dc (K=1024), folded to scalar
// ---------------------------------------------------------------------------
#define HT (96 * LSTR)  // buffer half: Ab(32x68) + Bc(64x68)

__global__ __launch_bounds__(256)
void tightness_kernel(const float* __restrict__ fn,
                      const float* __restrict__ centers,
                      const float* __restrict__ targets,
                      const float* __restrict__ sq,
                      const float* __restrict__ csq,
                      const float* __restrict__ validf,
                      const unsigned char* __restrict__ mask,
                      float* __restrict__ S_tight) {
  __shared__ __align__(16) float smem[2 * HT];
  const int tid  = threadIdx.x;
  const int wave = tid >> 5, lane = tid & 31;
  const int hf   = lane >> 4, l16 = lane & 15;
  const int wr   = wave >> 2, wc  = wave & 3;
  const int b0   = blockIdx.y * 32, c0 = blockIdx.x * 64;

  v8f acc = {};
  int buf = 0;
  stage_block(fn + (size_t)b0 * F_SZ, F_SZ, smem, LSTR, 32, 16);
  stage_block(centers + (size_t)c0 * F_SZ, F_SZ, smem + 32 * LSTR, LSTR, 64, 16);
  wait_async();
  __syncthreads();
  for (int k0 = 0; k0 < F_SZ; k0 += KTILE) {
    float* cur = smem + buf * HT;
    if (k0 + KTILE < F_SZ) {
      float* nxt = smem + (buf ^ 1) * HT;
      const int kn = k0 + KTILE;
      stage_block(fn + (size_t)b0 * F_SZ + kn, F_SZ, nxt, LSTR, 32, 16);
      stage_block(centers + (size_t)c0 * F_SZ + kn, F_SZ, nxt + 32 * LSTR, LSTR, 64, 16);
    }
    const float* Ab = cur;
    const float* Bc = cur + 32 * LSTR;
    #pragma unroll
    for (int kk = 0; kk < KTILE; kk += 4) {
      const v2f a = *(const v2f*)&Ab[(16 * wr + l16) * LSTR + kk + 2 * hf];
      const v2f b = *(const v2f*)&Bc[(16 * wc + l16) * LSTR + kk + 2 * hf];
      acc = wmma_f32(a, b, acc);
    }
    wait_async();
    __syncthreads();
    buf ^= 1;
  }

  float local = 0.f;
  #pragma unroll
  for (int v = 0; v < 8; ++v) {
    const int bi = b0 + 16 * wr + v + 8 * hf;
    const int dc = c0 + 16 * wc + l16;
    const float d2 = sq[bi] - 2.f * acc[v] + csq[dc];
    local += targets[(size_t)bi * DC_SZ + dc] * validf[dc] * d2 *
             (mask[bi] ? 1.f : 0.f);
  }
  local = waveReduceSum(local);
  if (lane == 0) atomicAdd(S_tight, local);
}

__global__ void finalize_kernel(const float* __restrict__ scal,
                                float* __restrict__ out) {
  const float Sdiv = scal[0], Stight = scal[1], M = scal[2];
  const float div   = -Sdiv / (M * (M - 1.f));
  const float tight = Stight / (M * (float)D_SZ);
  out[0] = 0.1f * div + 0.1f * tight;
  out[1] = div;
  out[2] = tight;
}

// ---------------------------------------------------------------------------
extern "C" void kernel_launch(void* const* d_in, const int* in_sizes, int n_in,
                              void* d_out, int out_size, void* d_ws, size_t ws_size,
                              hipStream_t stream) {
  const float* features = (const float*)d_in[0];
  const float* targets  = (const float*)d_in[1];
  const unsigned char* mask = (const unsigned char*)d_in[2]; // numpy bool = 1B

  float* ws      = (float*)d_ws;
  float* fn      = ws + OFF_FN;
  float* p       = ws + OFF_P;
  float* lp      = ws + OFF_LP;
  float* tm      = ws + OFF_TM;
  float* centers = ws + OFF_CENTERS;
  float* sq      = ws + OFF_SQ;
  float* E       = ws + OFF_E;
  float* wsum    = ws + OFF_WSUM;
  float* csq     = ws + OFF_CSQ;
  float* validf  = ws + OFF_VALID;
  float* scal    = ws + OFF_SCAL;   // [S_div, S_tight, M, pad]

  zero_kernel<<<1, 1, 0, stream>>>(scal);
  normalize_kernel<<<B_SZ, 256, 0, stream>>>(features, fn, sq);
  prep_kl_kernel<<<B_SZ, 256, 0, stream>>>(targets, mask, p, lp, tm, E);
  wsum_kernel<<<1, 256, 0, stream>>>(tm, mask, wsum, &scal[2]);
  diversity_kernel<<<dim3(B_SZ / 64, B_SZ / 32), 256, 0, stream>>>(
      fn, p, lp, sq, E, mask, &scal[0]);
  centers_kernel<<<dim3(F_SZ / 64, DC_SZ / 32), 256, 0, stream>>>(tm, fn, centers);
  centers_norm_kernel<<<DC_SZ, 256, 0, stream>>>(centers, wsum, csq, validf);
  tightness_kernel<<<dim3(DC_SZ / 64, B_SZ / 32), 256, 0, stream>>>(
      fn, centers, targets, sq, csq, validf, mask, &scal[1]);
  finalize_kernel<<<1, 1, 0, stream>>>(scal, (float*)d_out);
}